// GlobalModel_84396107366555
// MI455X (gfx1250) — compile-verified
//
#include <hip/hip_runtime.h>

typedef float v8f __attribute__((ext_vector_type(8)));
typedef float v2f __attribute__((ext_vector_type(2)));

#define NUM_SEG 16384      // B in the reference
#define ACC_WAVES 4096     // waves for the accumulation pass

// ---------------------------------------------------------------------------
// Pass 0: zero the workspace accumulators (seg_sum[B*4] ++ cnt[B])
// ---------------------------------------------------------------------------
__global__ void zero_ws_kernel(float* __restrict__ ws, int n) {
    int i = blockIdx.x * blockDim.x + threadIdx.x;
    if (i < n) ws[i] = 0.0f;
}

// ---------------------------------------------------------------------------
// Pass 1: segment sum + counts.
// Each wave owns a contiguous chunk of rows; lanes stride by 32 inside the
// chunk so the float4 row loads are fully coalesced (global_load_b128).
// batch is sorted -> run-length accumulate per lane, flush atomics only when
// the segment id changes (~10x fewer atomics; accumulators live in L2).
// ---------------------------------------------------------------------------
__device__ __forceinline__ void flush_seg(float* __restrict__ seg,
                                          float* __restrict__ cnt,
                                          int cur, float4 a, float c) {
    if (cur >= 0) {
        atomicAdd(&seg[cur * 4 + 0], a.x);
        atomicAdd(&seg[cur * 4 + 1], a.y);
        atomicAdd(&seg[cur * 4 + 2], a.z);
        atomicAdd(&seg[cur * 4 + 3], a.w);
        atomicAdd(&cnt[cur], c);
    }
}

__global__ void __launch_bounds__(256)
seg_accum_kernel(const float4* __restrict__ x, const int* __restrict__ batch,
                 float* __restrict__ seg, float* __restrict__ cnt,
                 int n, int chunk) {
    const int wave = (blockIdx.x * blockDim.x + threadIdx.x) >> 5;
    const int lane = threadIdx.x & 31;
    long long start = (long long)wave * (long long)chunk;
    long long end   = start + chunk;
    if (end > n) end = n;

    int cur = -1;
    float4 acc = make_float4(0.f, 0.f, 0.f, 0.f);
    float  c   = 0.f;

    for (long long i = start + lane; i < end; i += 32) {
        int    b = batch[i];
        float4 v = x[i];
        if (b != cur) {
            flush_seg(seg, cnt, cur, acc, c);
            cur = b; acc = v; c = 1.f;
        } else {
            acc.x += v.x; acc.y += v.y; acc.z += v.z; acc.w += v.w; c += 1.f;
        }
    }
    flush_seg(seg, cnt, cur, acc, c);
}

// ---------------------------------------------------------------------------
// Pass 2: per-segment mean, concat u, MLP via V_WMMA_F32_16X16X4_F32.
// One wave handles 16 graphs (rows). h (16x5, K-padded to 8) is the A matrix,
// W1 (5x5, zero-padded) is the B matrix; two K=4 WMMA steps give G = h@W1 in
// fp32 with the reference's exact precision. Bias + leaky-relu in VALU, then
// layer 2 (act @ W2[5x1]) folds into a per-lane multiply + 16-lane shfl_xor
// reduction over the D-matrix columns. No partial waves, no divergence at the
// WMMA -> EXEC is all ones as required.
//
// ISA layouts used (CDNA5 ISA):
//   A 16x4 f32: lanes 0-15 = rows M, VGPR0/1 = K0/K1; lanes 16-31 = K2/K3.
//   B 4x16 f32: VGPR v holds K-rows v (lanes 0-15) and v+2 (lanes 16-31).
//   C/D 16x16:  VGPR j holds row j (lanes 0-15) and row j+8 (lanes 16-31).
// ---------------------------------------------------------------------------
__global__ void __launch_bounds__(256)
mlp_wmma_kernel(const float* __restrict__ seg, const float* __restrict__ cnt,
                const float* __restrict__ u,   const float* __restrict__ W1,
                const float* __restrict__ b1,  const float* __restrict__ W2,
                const float* __restrict__ b2,  float* __restrict__ out) {
    const int wave = (blockIdx.x * blockDim.x + threadIdx.x) >> 5; // 0..1023
    const int lane = threadIdx.x & 31;
    const int half = lane >> 4;      // 0: lanes 0-15, 1: lanes 16-31
    const int n    = lane & 15;      // column index (D) / row index (A load)
    const int g0   = wave * 16;
    const int g    = g0 + n;         // graph this lane loads A-row data for

    float cv  = cnt[g];
    float inv = (cv > 0.f) ? (1.f / cv) : 0.f;

    // A-matrix fragments: h[g] = [u, mean0, mean1, mean2, mean3, 0, 0, 0]
    v2f a0, a1;
    if (half == 0) {                       // K = 0,1  (step0) / K = 4,5 (step1)
        a0.x = u[g];
        a0.y = seg[g * 4 + 0] * inv;
        a1.x = seg[g * 4 + 3] * inv;       // K=4 -> mean3
        a1.y = 0.f;
    } else {                               // K = 2,3  (step0) / K = 6,7 (step1)
        a0.x = seg[g * 4 + 1] * inv;
        a0.y = seg[g * 4 + 2] * inv;
        a1.x = 0.f;
        a1.y = 0.f;
    }

    // B-matrix fragments: W1 is (5x5) row-major [k][n], zero-padded to 8x16.
    float w1_0 = (n < 5) ? W1[0 * 5 + n] : 0.f;
    float w1_1 = (n < 5) ? W1[1 * 5 + n] : 0.f;
    float w1_2 = (n < 5) ? W1[2 * 5 + n] : 0.f;
    float w1_3 = (n < 5) ? W1[3 * 5 + n] : 0.f;
    float w1_4 = (n < 5) ? W1[4 * 5 + n] : 0.f;
    v2f bm0, bm1;
    bm0.x = (half == 0) ? w1_0 : w1_2;     // step0: K-rows 0/2
    bm0.y = (half == 0) ? w1_1 : w1_3;     // step0: K-rows 1/3
    bm1.x = (half == 0) ? w1_4 : 0.f;      // step1: K-row 4 (rows 5..7 zero)
    bm1.y = 0.f;

    v8f acc = {};
    acc = __builtin_amdgcn_wmma_f32_16x16x4_f32(false, a0, false, bm0,
                                                (short)0, acc, false, false);
    acc = __builtin_amdgcn_wmma_f32_16x16x4_f32(false, a1, false, bm1,
                                                (short)0, acc, false, false);

    // Layer 1 epilogue + layer 2 (act @ W2 + b2), fused per column.
    float bias = (n < 5) ? b1[n] : 0.f;
    float w2n  = (n < 5) ? W2[n] : 0.f;

    float red[8];
#pragma unroll
    for (int j = 0; j < 8; ++j) {
        float gv  = acc[j] + bias;
        float act = (gv > 0.f) ? gv : 0.1f * gv;   // leaky_relu(0.1)
        float p   = act * w2n;
        p += __shfl_xor(p, 1, 32);                 // reduce within 16-lane half
        p += __shfl_xor(p, 2, 32);
        p += __shfl_xor(p, 4, 32);
        p += __shfl_xor(p, 8, 32);
        red[j] = p;                                // row (j + half*8) result
    }

    if (n == 0) {
        float bb = b2[0];
#pragma unroll
        for (int j = 0; j < 8; ++j)
            out[g0 + half * 8 + j] = red[j] + bb;
    }
}

// ---------------------------------------------------------------------------
extern "C" void kernel_launch(void* const* d_in, const int* in_sizes, int n_in,
                              void* d_out, int out_size, void* d_ws, size_t ws_size,
                              hipStream_t stream) {
    const float4* x     = (const float4*)d_in[0];   // [N,4] fp32, row = 16B
    const int*    batch = (const int*)  d_in[1];    // [N] int
    const float*  u     = (const float*)d_in[2];    // [B,1]
    const float*  W1    = (const float*)d_in[3];    // [5,5]
    const float*  b1    = (const float*)d_in[4];    // [5]
    const float*  W2    = (const float*)d_in[5];    // [5,1]
    const float*  b2    = (const float*)d_in[6];    // [1]
    float*        out   = (float*)d_out;            // [B,1]

    const int N = in_sizes[1];                      // number of nodes

    float* seg = (float*)d_ws;                      // [B*4]
    float* cnt = seg + NUM_SEG * 4;                 // [B]

    // Pass 0: zero accumulators (B*5 floats)
    {
        int n0 = NUM_SEG * 5;
        zero_ws_kernel<<<(n0 + 255) / 256, 256, 0, stream>>>(seg, n0);
    }

    // Pass 1: streaming segment sum (memory-bound, ~100 MB total traffic)
    {
        int chunk  = (N + ACC_WAVES - 1) / ACC_WAVES;
        int blocks = (ACC_WAVES * 32) / 256;        // 512 blocks x 256 threads
        seg_accum_kernel<<<blocks, 256, 0, stream>>>(x, batch, seg, cnt, N, chunk);
    }

    // Pass 2: mean + MLP via fp32 WMMA. 1024 waves = 128 blocks x 256 threads.
    {
        int blocks = (NUM_SEG / 16) * 32 / 256;     // 128
        mlp_wmma_kernel<<<blocks, 256, 0, stream>>>(seg, cnt, u, W1, b1, W2, b2, out);
    }
}